// SageAttnBinary_80796924772719
// MI455X (gfx1250) — compile-verified
//
#include <hip/hip_runtime.h>
#include <math.h>

typedef float v2f __attribute__((ext_vector_type(2)));
typedef float v8f __attribute__((ext_vector_type(8)));

#define HID 128
#define ROWS_PER_BLOCK 64
#define LDS_STRIDE 132   // 132 mod 64 == 4 -> conflict-free row-indexed ds_load_b64

__device__ __forceinline__ float gelu_erf(float v) {
    return 0.5f * v * (1.0f + erff(v * 0.70710678118654752f));
}

// ---------------------------------------------------------------- degree
__global__ void deg_kernel(const int* __restrict__ dst, float* __restrict__ cnt, int n_edges) {
    int e = blockIdx.x * blockDim.x + threadIdx.x;
    if (e < n_edges) atomicAdd(&cnt[dst[e]], 1.0f);
}

__global__ void invcnt_kernel(float* __restrict__ cnt, int n_nodes) {
    int i = blockIdx.x * blockDim.x + threadIdx.x;
    if (i < n_nodes) cnt[i] = 1.0f / fmaxf(cnt[i], 1.0f);
}

// ------------------------------------------------------- edge scatter-add
// One wave (32 lanes) per edge: coalesced 512B row read, 4 f32 atomics/lane.
// agg buffer (51.2 MB) is L2-resident on MI455X (192 MB L2).
__global__ __launch_bounds__(256) void scatter_kernel(
    const float* __restrict__ feat, const int* __restrict__ src,
    const int* __restrict__ dst, float* __restrict__ agg, int n_edges) {
    long long gid = (long long)blockIdx.x * blockDim.x + threadIdx.x;
    int e = (int)(gid >> 5);
    int lane = (int)(gid & 31);
    if (e >= n_edges) return;
    int s = src[e], d = dst[e];
    const float4 v = *(const float4*)(feat + (size_t)s * HID + lane * 4);
    float* out = agg + (size_t)d * HID + lane * 4;
    atomicAdd(out + 0, v.x);
    atomicAdd(out + 1, v.y);
    atomicAdd(out + 2, v.z);
    atomicAdd(out + 3, v.w);
}

// ------------------------------------------------------- fused WMMA GEMM
// out = gelu( (A .* inv?) @ Wl + bl [+ X @ Wr] ), A,X: [N,128], W: [128,128]
// Block: 256 threads = 8 waves, owns 64 rows (4 row-tiles of 16).
// Wave w owns cols [16w,16w+16). Weight fragments are preloaded into
// registers (64 VGPRs per matrix per wave) and reused across 4 row-tiles,
// so the inner loop is ds_load_b64 (A) + v_wmma_f32_16x16x4_f32 only.
template <bool DUAL, bool INV>
__global__ __launch_bounds__(256) void sage_gemm_kernel(
    const float* __restrict__ A, const float* __restrict__ invcnt,
    const float* __restrict__ X, const float* __restrict__ Wl,
    const float* __restrict__ bl, const float* __restrict__ Wr,
    float* __restrict__ out, int n_nodes) {
    __shared__ float sA[ROWS_PER_BLOCK][LDS_STRIDE];
    __shared__ float sX[DUAL ? ROWS_PER_BLOCK : 1][LDS_STRIDE];
    const int row0 = blockIdx.x * ROWS_PER_BLOCK;
    const int tid = threadIdx.x;

    // ---- stage A (and X) rows into LDS: float4 global reads, scalar LDS writes
    for (int i = tid; i < ROWS_PER_BLOCK * (HID / 4); i += 256) {
        int r = i >> 5;             // row within block
        int c4 = (i & 31) * 4;      // starting column of float4
        int gr = row0 + r;
        float4 a = {0.f, 0.f, 0.f, 0.f}, x = {0.f, 0.f, 0.f, 0.f};
        if (gr < n_nodes) {
            a = *(const float4*)(A + (size_t)gr * HID + c4);
            if (INV) {
                float ic = invcnt[gr];
                a.x *= ic; a.y *= ic; a.z *= ic; a.w *= ic;
            }
            if (DUAL) x = *(const float4*)(X + (size_t)gr * HID + c4);
        }
        sA[r][c4 + 0] = a.x; sA[r][c4 + 1] = a.y;
        sA[r][c4 + 2] = a.z; sA[r][c4 + 3] = a.w;
        if (DUAL) {
            sX[r][c4 + 0] = x.x; sX[r][c4 + 1] = x.y;
            sX[r][c4 + 2] = x.z; sX[r][c4 + 3] = x.w;
        }
    }
    __syncthreads();

    const int wave = tid >> 5;       // N-tile index 0..7
    const int lane = tid & 31;
    const int n0 = wave * 16;
    const int lm = lane & 15;        // row (A) / col (B,D) within tile
    const int hi = lane >> 4;        // selects K pair {0,1} vs {2,3}
    const int kh = 2 * hi;

    // ---- preload weight fragments into registers (reused for 4 row-tiles)
    v2f bL[32];
    v2f bR[32];
#pragma unroll
    for (int kk = 0; kk < 32; kk++) {
        int k = kk * 4 + kh;
        bL[kk].x = Wl[k * HID + n0 + lm];
        bL[kk].y = Wl[(k + 1) * HID + n0 + lm];
        if (DUAL) {
            bR[kk].x = Wr[k * HID + n0 + lm];
            bR[kk].y = Wr[(k + 1) * HID + n0 + lm];
        }
    }

    const float bias = bl[n0 + lm];

#pragma unroll
    for (int rt = 0; rt < 4; rt++) {
        const int r0 = rt * 16;
        v8f c = {};
#pragma unroll
        for (int kk = 0; kk < 32; kk++) {
            int k = kk * 4;
            v2f a;
            a.x = sA[r0 + lm][k + kh];
            a.y = sA[r0 + lm][k + kh + 1];
            c = __builtin_amdgcn_wmma_f32_16x16x4_f32(false, a, false, bL[kk],
                                                      (short)0, c, false, false);
            if (DUAL) {
                v2f a2;
                a2.x = sX[r0 + lm][k + kh];
                a2.y = sX[r0 + lm][k + kh + 1];
                c = __builtin_amdgcn_wmma_f32_16x16x4_f32(false, a2, false, bR[kk],
                                                          (short)0, c, false, false);
            }
        }
#pragma unroll
        for (int j = 0; j < 8; j++) {
            int r = r0 + hi * 8 + j;   // D layout: VGPR j = row j (lanes<16) / 8+j
            int gr = row0 + r;
            if (gr < n_nodes) {
                out[(size_t)gr * HID + n0 + lm] = gelu_erf(c[j] + bias);
            }
        }
    }
}

// ---------------------------------------------------- gate: tmp @ Wg2 + b
__global__ __launch_bounds__(256) void gate_reduce_kernel(
    const float* __restrict__ tmp, const float* __restrict__ Wg2,
    const float* __restrict__ bg2, float* __restrict__ gate, int n_nodes) {
    int wave = threadIdx.x >> 5, lane = threadIdx.x & 31;
    int n = blockIdx.x * 8 + wave;
    if (n >= n_nodes) return;
    float v = 0.0f;
#pragma unroll
    for (int i = 0; i < 4; i++) {
        int col = lane + 32 * i;
        v += tmp[(size_t)n * HID + col] * Wg2[col];
    }
    for (int off = 16; off > 0; off >>= 1) v += __shfl_down(v, off, 32);
    if (lane == 0) gate[n] = v + bg2[0];
}

// ------------------------------------------------ segment max via atomics
__device__ __forceinline__ unsigned flip_f32(float f) {
    unsigned u = __float_as_uint(f);
    return (u & 0x80000000u) ? ~u : (u | 0x80000000u);
}
__device__ __forceinline__ float unflip_f32(unsigned u) {
    return (u & 0x80000000u) ? __uint_as_float(u ^ 0x80000000u)
                             : __uint_as_float(~u);
}

__global__ void gmax_init_kernel(unsigned* __restrict__ gmax, int n_graphs) {
    int i = blockIdx.x * blockDim.x + threadIdx.x;
    if (i < n_graphs) gmax[i] = 0x007FFFFFu;  // flip(-inf)
}

__global__ void gmax_kernel(const float* __restrict__ gate,
                            const int* __restrict__ batch,
                            unsigned* __restrict__ gmax, int n_nodes) {
    int n = blockIdx.x * blockDim.x + threadIdx.x;
    if (n < n_nodes) atomicMax(&gmax[batch[n]], flip_f32(gate[n]));
}

__global__ void exp_kernel(const float* __restrict__ gate,
                           const int* __restrict__ batch,
                           const unsigned* __restrict__ gmax,
                           float* __restrict__ ebuf,
                           float* __restrict__ denom, int n_nodes) {
    int n = blockIdx.x * blockDim.x + threadIdx.x;
    if (n >= n_nodes) return;
    int g = batch[n];
    float e = expf(gate[n] - unflip_f32(gmax[g]));
    ebuf[n] = e;
    atomicAdd(&denom[g], e);
}

// --------------------------- pooling: exploit sorted batch, few atomics
__global__ __launch_bounds__(128) void pool_kernel(
    const float* __restrict__ h, const float* __restrict__ ebuf,
    const float* __restrict__ denom, const int* __restrict__ batch,
    float* __restrict__ gacc, int n_nodes) {
    int t = threadIdx.x;
    int base = blockIdx.x * 16;
    if (base >= n_nodes) return;
    float acc = 0.0f;
    int curg = batch[base];
    for (int i = 0; i < 16; i++) {
        int n = base + i;
        if (n >= n_nodes) break;
        int g = batch[n];
        if (g != curg) {
            atomicAdd(&gacc[curg * HID + t], acc);
            acc = 0.0f;
            curg = g;
        }
        float coef = ebuf[n] / denom[g];
        acc += coef * h[(size_t)n * HID + t];
    }
    atomicAdd(&gacc[curg * HID + t], acc);
}

// ------------------------------------------------------------------ head
__global__ __launch_bounds__(128) void head_kernel(
    const float* __restrict__ gacc, const float* __restrict__ Wh,
    const float* __restrict__ bh, float* __restrict__ out) {
    __shared__ float red[4];
    int g = blockIdx.x, t = threadIdx.x;
    float v = gacc[g * HID + t] * Wh[t];
    for (int off = 16; off > 0; off >>= 1) v += __shfl_down(v, off, 32);
    if ((t & 31) == 0) red[t >> 5] = v;
    __syncthreads();
    if (t == 0) out[g] = red[0] + red[1] + red[2] + red[3] + bh[0];
}

// ==================================================================
extern "C" void kernel_launch(void* const* d_in, const int* in_sizes, int n_in,
                              void* d_out, int out_size, void* d_ws, size_t ws_size,
                              hipStream_t stream) {
    const float* x    = (const float*)d_in[0];
    const int*   ei   = (const int*)d_in[1];
    const int*   batch= (const int*)d_in[2];
    const float* Wl0  = (const float*)d_in[3];
    const float* bl0  = (const float*)d_in[4];
    const float* Wr0  = (const float*)d_in[5];
    const float* Wl1  = (const float*)d_in[6];
    const float* bl1  = (const float*)d_in[7];
    const float* Wr1  = (const float*)d_in[8];
    const float* Wg1  = (const float*)d_in[9];
    const float* bg1  = (const float*)d_in[10];
    const float* Wg2  = (const float*)d_in[11];
    const float* bg2  = (const float*)d_in[12];
    const float* Wh   = (const float*)d_in[13];
    const float* bh   = (const float*)d_in[14];
    float* out = (float*)d_out;

    const int N = in_sizes[0] / HID;      // 100000
    const int E = in_sizes[1] / 2;        // 1600000
    const int G = out_size;               // 64
    const int* src = ei;
    const int* dst = ei + E;

    // workspace carve-out
    size_t off = 0;
    auto carve = [&](size_t bytes) -> void* {
        off = (off + 255) & ~(size_t)255;
        void* p = (char*)d_ws + off;
        off += bytes;
        return p;
    };
    const size_t NB = (size_t)N * HID * sizeof(float);
    float*    agg   = (float*)carve(NB);           // also reused as gate-MLP tmp
    float*    h0    = (float*)carve(NB);
    float*    h1    = (float*)carve(NB);
    float*    cnt   = (float*)carve((size_t)N * sizeof(float));
    float*    gate  = (float*)carve((size_t)N * sizeof(float));
    float*    ebuf  = (float*)carve((size_t)N * sizeof(float));
    float*    denom = (float*)carve((size_t)G * sizeof(float));
    unsigned* gmax  = (unsigned*)carve((size_t)G * sizeof(unsigned));
    float*    gacc  = (float*)carve((size_t)G * HID * sizeof(float));
    (void)ws_size; (void)n_in;

    const int TB = 256;
    const int gemm_blocks    = (N + ROWS_PER_BLOCK - 1) / ROWS_PER_BLOCK;
    const int scatter_blocks = (int)(((long long)E * 32 + TB - 1) / TB);

    // degree + reciprocal
    hipMemsetAsync(cnt, 0, (size_t)N * sizeof(float), stream);
    deg_kernel<<<(E + TB - 1) / TB, TB, 0, stream>>>(dst, cnt, E);
    invcnt_kernel<<<(N + TB - 1) / TB, TB, 0, stream>>>(cnt, N);

    // SAGE layer 0
    hipMemsetAsync(agg, 0, NB, stream);
    scatter_kernel<<<scatter_blocks, TB, 0, stream>>>(x, src, dst, agg, E);
    sage_gemm_kernel<true, true><<<gemm_blocks, TB, 0, stream>>>(
        agg, cnt, x, Wl0, bl0, Wr0, h0, N);

    // SAGE layer 1
    hipMemsetAsync(agg, 0, NB, stream);
    scatter_kernel<<<scatter_blocks, TB, 0, stream>>>(h0, src, dst, agg, E);
    sage_gemm_kernel<true, true><<<gemm_blocks, TB, 0, stream>>>(
        agg, cnt, h0, Wl1, bl1, Wr1, h1, N);

    // gate MLP hidden: tmp = gelu(h1 @ Wg1 + bg1)  (reuse agg buffer)
    sage_gemm_kernel<false, false><<<gemm_blocks, TB, 0, stream>>>(
        h1, nullptr, nullptr, Wg1, bg1, nullptr, agg, N);
    gate_reduce_kernel<<<(N + 7) / 8, TB, 0, stream>>>(agg, Wg2, bg2, gate, N);

    // per-graph softmax
    gmax_init_kernel<<<1, 64, 0, stream>>>(gmax, G);
    gmax_kernel<<<(N + TB - 1) / TB, TB, 0, stream>>>(gate, batch, gmax, N);
    hipMemsetAsync(denom, 0, (size_t)G * sizeof(float), stream);
    exp_kernel<<<(N + TB - 1) / TB, TB, 0, stream>>>(gate, batch, gmax, ebuf, denom, N);

    // attention pooling + head
    hipMemsetAsync(gacc, 0, (size_t)G * HID * sizeof(float), stream);
    pool_kernel<<<(N + 15) / 16, 128, 0, stream>>>(h1, ebuf, denom, batch, gacc, N);
    head_kernel<<<G, 128, 0, stream>>>(gacc, Wh, bh, out);
}